// HebbianMemory_37220186588012
// MI455X (gfx1250) — compile-verified
//
#include <hip/hip_runtime.h>

// Hebbian linear-attention forward for MI455X (gfx1250, wave32).
// Chunked linear attention: all heavy math routed through V_WMMA_F32_16X16X4_F32
// (fp32 in / fp32 acc -> matches the fp32 reference numerics).
// Global->LDS staging uses GLOBAL_LOAD_ASYNC_TO_LDS_B128 (ASYNCcnt) with
// double buffering in the GEMMs when the toolchain exposes the builtins.

#define DM 512           // d_model
#define NH 8             // heads
#define EH 64            // head dim
#define NB 2             // batch
#define SLEN 4096        // sequence length
#define CK 64            // chunk length
#define NCH (SLEN / CK)  // 64 chunks
#define NTOK (NB * SLEN) // 8192 tokens
#define D3 (3 * DM)      // 1536

#if defined(__has_builtin)
#if __has_builtin(__builtin_amdgcn_global_load_async_to_lds_b128)
#define HAVE_ASYNC_LDS 1
#endif
#endif
#ifndef HAVE_ASYNC_LDS
#define HAVE_ASYNC_LDS 0
#endif

#if HAVE_ASYNC_LDS
// Builtin prototype (from clang diagnostic): first arg is AS1 pointer to a
// 16-byte int vector; second is the LDS-side pointer. 16-byte async copy
// global -> LDS (per-lane addresses, tracked by ASYNCcnt).
typedef int v4i_vs __attribute__((vector_size(4 * sizeof(int))));
#define ASYNC_CP16(dst_lds, src_glb)                                          \
  __builtin_amdgcn_global_load_async_to_lds_b128(                             \
      (__attribute__((address_space(1))) v4i_vs*)(src_glb),                   \
      (__attribute__((address_space(3))) v4i_vs*)(dst_lds), 0, 0)
#if __has_builtin(__builtin_amdgcn_s_wait_asynccnt)
#define WAIT_ASYNC(n) __builtin_amdgcn_s_wait_asynccnt(n)
#else
#define WAIT_ASYNC(n) asm volatile("s_wait_asynccnt " #n ::: "memory")
#endif
#define NBUF 2
#else
#define NBUF 1
#endif

typedef float v2f __attribute__((ext_vector_type(2)));
typedef float v8f __attribute__((ext_vector_type(8)));

__device__ __forceinline__ v8f zero_v8f() {
  v8f z = {0.f, 0.f, 0.f, 0.f, 0.f, 0.f, 0.f, 0.f};
  return z;
}

// One 16x16 f32 WMMA tile accumulating over kdim (multiple of 4).
// A element (m,k) at A[m*asM + k*asK]; B element (k,n) at B[k*bsK + n*bsN].
// Per-lane layout (ISA 7.12.2): A lanes 0-15 hold K={0,1}, lanes 16-31 K={2,3};
// B mirrored; C/D: VGPR i holds M=i (lanes 0-15) / M=i+8 (lanes 16-31), N=lane&15.
__device__ __forceinline__ v8f wmma_k(const float* A, int asM, int asK,
                                      const float* B, int bsK, int bsN,
                                      int kdim, v8f acc) {
  const int lane = threadIdx.x & 31;
  const int mn = lane & 15;
  const int kh = (lane >> 4) << 1; // 0 or 2
  for (int k0 = 0; k0 < kdim; k0 += 4) {
    const int ka = k0 + kh;
    v2f a, b;
    a.x = A[mn * asM + ka * asK];
    a.y = A[mn * asM + (ka + 1) * asK];
    b.x = B[ka * bsK + mn * bsN];
    b.y = B[(ka + 1) * bsK + mn * bsN];
    acc = __builtin_amdgcn_wmma_f32_16x16x4_f32(false, a, false, b, (short)0,
                                                acc, false, false);
  }
  return acc;
}

// ---------------- Kernel 1: QKV GEMM + feature-map epilogue ----------------
// qkv = x @ qkv_w^T + b ; q' = (elu(q)+1)/8 ; k' = elu(k)+1 ; v raw.
// Outputs in [b][h][l][e] layout for chunked access.
__global__ __launch_bounds__(256) void k_qkv(
    const float* __restrict__ x, const float* __restrict__ w,
    const float* __restrict__ bias, float* __restrict__ qp,
    float* __restrict__ kp, float* __restrict__ vp) {
  __shared__ float lA[NBUF][64][65];
  __shared__ float lB[NBUF][64][65]; // W rows [n][k] -> read as B^T
  const int row0 = blockIdx.x * 64;
  const int col0 = blockIdx.y * 64;
  const int wid = threadIdx.x >> 5;
  v8f acc[2];
  acc[0] = zero_v8f();
  acc[1] = zero_v8f();
#if HAVE_ASYNC_LDS
  for (int i = threadIdx.x; i < 64 * 16; i += 256) {
    const int r = i >> 4;
    const int c = (i & 15) << 2;
    ASYNC_CP16(&lA[0][r][c], &x[(size_t)(row0 + r) * DM + c]);
    ASYNC_CP16(&lB[0][r][c], &w[(size_t)(col0 + r) * DM + c]);
  }
#endif
  int buf = 0;
  for (int k0 = 0; k0 < DM; k0 += 64) {
#if HAVE_ASYNC_LDS
    if (k0 + 64 < DM) {
      const int nb = buf ^ 1;
      for (int i = threadIdx.x; i < 64 * 16; i += 256) {
        const int r = i >> 4;
        const int c = (i & 15) << 2;
        ASYNC_CP16(&lA[nb][r][c], &x[(size_t)(row0 + r) * DM + k0 + 64 + c]);
        ASYNC_CP16(&lB[nb][r][c], &w[(size_t)(col0 + r) * DM + k0 + 64 + c]);
      }
      WAIT_ASYNC(8); // retire the 8 older per-thread copies (current buffer)
    } else {
      WAIT_ASYNC(0);
    }
    __syncthreads();
#else
    __syncthreads();
    for (int i = threadIdx.x; i < 64 * 16; i += 256) {
      const int r = i >> 4;
      const int c = (i & 15) << 2;
      const float4 va = *(const float4*)&x[(size_t)(row0 + r) * DM + k0 + c];
      lA[0][r][c] = va.x; lA[0][r][c + 1] = va.y;
      lA[0][r][c + 2] = va.z; lA[0][r][c + 3] = va.w;
      const float4 vb = *(const float4*)&w[(size_t)(col0 + r) * DM + k0 + c];
      lB[0][r][c] = vb.x; lB[0][r][c + 1] = vb.y;
      lB[0][r][c + 2] = vb.z; lB[0][r][c + 3] = vb.w;
    }
    __syncthreads();
#endif
#pragma unroll
    for (int t = 0; t < 2; ++t) {
      const int t16 = wid * 2 + t;
      const int ti = t16 >> 2, tj = t16 & 3;
      acc[t] = wmma_k(&lA[buf][ti * 16][0], 65, 1, &lB[buf][tj * 16][0], 1, 65,
                      64, acc[t]);
    }
#if HAVE_ASYNC_LDS
    __syncthreads(); // everyone done reading buf before it is restaged
    buf ^= 1;
#endif
  }
  const int lane = threadIdx.x & 31;
  const int col = lane & 15;
  const int rb = (lane >> 4) << 3;
#pragma unroll
  for (int t = 0; t < 2; ++t) {
    const int t16 = wid * 2 + t;
    const int ti = t16 >> 2, tj = t16 & 3;
#pragma unroll
    for (int i = 0; i < 8; ++i) {
      const int gr = row0 + ti * 16 + rb + i; // token
      const int gc = col0 + tj * 16 + col;    // 0..1535
      const float val = acc[t][i] + bias[gc];
      const int bb = gr >> 12;      // / SLEN
      const int l = gr & (SLEN - 1);
      const int sect = gc >> 9;     // 0=q 1=k 2=v
      const int d = gc & 511;
      const int h = d >> 6, e = d & 63;
      const size_t oidx = (((size_t)bb * NH + h) * SLEN + l) * EH + e;
      if (sect == 0)
        qp[oidx] = (val > 0.f ? val + 1.f : __expf(val)) * 0.125f; // /sqrt(E)
      else if (sect == 1)
        kp[oidx] = (val > 0.f ? val + 1.f : __expf(val));
      else
        vp[oidx] = val;
    }
  }
}

// ---------------- Kernel 2: decay logits + chunk-relative cumsum -----------
__global__ __launch_bounds__(64) void k_decay(
    const float* __restrict__ x, const float* __restrict__ dw,
    const float* __restrict__ db, float* __restrict__ cumul,
    float* __restrict__ csum) {
  __shared__ float sw[NH * DM];
  __shared__ float ll[CK][NH];
  const int b = blockIdx.x / NCH;
  const int ch = blockIdx.x % NCH;
  const int t = threadIdx.x;
  for (int i = t; i < NH * DM; i += 64) sw[i] = dw[i];
  __syncthreads();
  const float* xr = &x[((size_t)b * SLEN + ch * CK + t) * DM];
  float dots[NH];
#pragma unroll
  for (int h = 0; h < NH; ++h) dots[h] = db[h];
  for (int k = 0; k < DM; ++k) {
    const float xv = xr[k];
#pragma unroll
    for (int h = 0; h < NH; ++h) dots[h] += xv * sw[h * DM + k];
  }
#pragma unroll
  for (int h = 0; h < NH; ++h) {
    const float sig = 1.f / (1.f + __expf(-dots[h]));
    ll[t][h] = __logf(0.9f + 0.1f * sig); // log lambda (clip never active here)
  }
  __syncthreads();
  if (t < NH) {
    float c = 0.f;
    const size_t base = ((size_t)b * NH + t) * SLEN + ch * CK;
    for (int s = 0; s < CK; ++s) {
      c += ll[s][t];
      cumul[base + s] = c; // inclusive chunk-relative cumsum
    }
    csum[((size_t)b * NH + t) * NCH + ch] = c; // chunk total
  }
}

// ---------------- Kernel 3: per-chunk local state S_loc = (K.w)^T V --------
__global__ __launch_bounds__(256) void k_chunk_local(
    const float* __restrict__ kp, const float* __restrict__ vp,
    const float* __restrict__ cumul, const float* __restrict__ csum,
    float* __restrict__ sloc, float* __restrict__ ksl) {
  __shared__ float sK[CK][EH + 1];
  __shared__ float sV[CK][EH + 1];
  __shared__ float swd[CK];
  const int ch = blockIdx.x % NCH;
  const int bh = blockIdx.x / NCH;
  const size_t base = ((size_t)bh * SLEN + ch * CK) * EH;
  if (threadIdx.x < CK) {
    const float cend = csum[(size_t)bh * NCH + ch];
    swd[threadIdx.x] =
        __expf(cend - cumul[(size_t)bh * SLEN + ch * CK + threadIdx.x]);
  }
  __syncthreads();
  for (int i = threadIdx.x; i < CK * EH / 4; i += 256) {
    const int r = i >> 4;
    const int c = (i & 15) << 2;
#if HAVE_ASYNC_LDS
    ASYNC_CP16(&sV[r][c], &vp[base + (size_t)r * EH + c]); // raw copy: async
#else
    const float4 v4 = *(const float4*)&vp[base + (size_t)r * EH + c];
    sV[r][c] = v4.x; sV[r][c + 1] = v4.y; sV[r][c + 2] = v4.z; sV[r][c + 3] = v4.w;
#endif
    const float wd = swd[r];
    const float4 k4 = *(const float4*)&kp[base + (size_t)r * EH + c];
    sK[r][c] = k4.x * wd; sK[r][c + 1] = k4.y * wd;
    sK[r][c + 2] = k4.z * wd; sK[r][c + 3] = k4.w * wd;
  }
#if HAVE_ASYNC_LDS
  WAIT_ASYNC(0);
#endif
  __syncthreads();
  const int wid = threadIdx.x >> 5;
  const int lane = threadIdx.x & 31;
  const int col = lane & 15;
  const int rb = (lane >> 4) << 3;
  const size_t off = ((size_t)bh * NCH + ch) * EH * EH;
#pragma unroll
  for (int t = 0; t < 2; ++t) {
    const int t16 = wid * 2 + t;
    const int ti = t16 >> 2, tj = t16 & 3;
    // S[e][f] = sum_s sK[s][e]*sV[s][f]  (A read transposed from LDS)
    v8f acc = zero_v8f();
    acc = wmma_k(&sK[0][ti * 16], 1, EH + 1, &sV[0][tj * 16], EH + 1, 1, CK, acc);
#pragma unroll
    for (int i = 0; i < 8; ++i)
      sloc[off + (size_t)(ti * 16 + rb + i) * EH + tj * 16 + col] = acc[i];
  }
  if (threadIdx.x < EH) {
    float s = 0.f;
    for (int r = 0; r < CK; ++r) s += sK[r][threadIdx.x];
    ksl[((size_t)bh * NCH + ch) * EH + threadIdx.x] = s;
  }
}

// ---------------- Kernel 4: sequential chunk-level scan (in place) ---------
// Converts sloc/ksl into *exclusive prefixes* (state before each chunk).
__global__ __launch_bounds__(256) void k_scan(float* __restrict__ sloc,
                                              float* __restrict__ ksl,
                                              const float* __restrict__ csum) {
  const int bh = blockIdx.x;
  const int tid = threadIdx.x;
  float s[16];
#pragma unroll
  for (int j = 0; j < 16; ++j) s[j] = 0.f;
  float ka = 0.f;
  for (int ch = 0; ch < NCH; ++ch) {
    const size_t off = ((size_t)bh * NCH + ch) * EH * EH;
    const float g = __expf(csum[(size_t)bh * NCH + ch]);
#pragma unroll
    for (int j = 0; j < 16; ++j) {
      const size_t p = off + (size_t)j * 256 + tid;
      const float tmp = sloc[p];
      sloc[p] = s[j];            // exclusive prefix
      s[j] = g * s[j] + tmp;     // carry
    }
    if (tid < EH) {
      const size_t p = ((size_t)bh * NCH + ch) * EH + tid;
      const float tmp = ksl[p];
      ksl[p] = ka;
      ka = g * ka + tmp;
    }
  }
}

// ---------------- Kernel 5: chunk output (intra + inter) -------------------
__global__ __launch_bounds__(256) void k_chunk_out(
    const float* __restrict__ qp, const float* __restrict__ kp,
    const float* __restrict__ vp, const float* __restrict__ cumul,
    const float* __restrict__ spre, const float* __restrict__ kpre,
    float* __restrict__ y) {
  __shared__ float sQ[CK][EH + 1];   // q * exp(+c_t)
  __shared__ float sKd[CK][EH + 1];  // k * exp(-c_s)
  __shared__ float sV[CK][EH + 1];
  __shared__ float sS[EH][EH + 1];   // prefix state
  __shared__ float sA[CK][CK + 1];   // masked attention
  __shared__ float seq[CK], sek[CK], skp[EH], sden[CK];
  const int ch = blockIdx.x % NCH;
  const int bh = blockIdx.x / NCH;
  const int b = bh / NH, h = bh % NH;
  const size_t base = ((size_t)bh * SLEN + ch * CK) * EH;
  const size_t soff = ((size_t)bh * NCH + ch) * EH * EH;
#if HAVE_ASYNC_LDS
  // Raw tensors can start streaming into LDS immediately (ASYNCcnt path).
  for (int i = threadIdx.x; i < CK * EH / 4; i += 256) {
    const int r = i >> 4;
    const int c = (i & 15) << 2;
    ASYNC_CP16(&sV[r][c], &vp[base + (size_t)r * EH + c]);
    ASYNC_CP16(&sS[r][c], &spre[soff + (size_t)r * EH + c]);
  }
#endif
  if (threadIdx.x < CK) {
    const float cu = cumul[(size_t)bh * SLEN + ch * CK + threadIdx.x];
    seq[threadIdx.x] = __expf(cu);
    sek[threadIdx.x] = __expf(-cu);
  } else if (threadIdx.x < CK + EH) {
    const int e = threadIdx.x - CK;
    skp[e] = kpre[((size_t)bh * NCH + ch) * EH + e];
  }
  __syncthreads();
  for (int i = threadIdx.x; i < CK * EH / 4; i += 256) {
    const int r = i >> 4;
    const int c = (i & 15) << 2;
    const float eq = seq[r], ek = sek[r];
    const float4 q4 = *(const float4*)&qp[base + (size_t)r * EH + c];
    sQ[r][c] = q4.x * eq; sQ[r][c + 1] = q4.y * eq;
    sQ[r][c + 2] = q4.z * eq; sQ[r][c + 3] = q4.w * eq;
    const float4 k4 = *(const float4*)&kp[base + (size_t)r * EH + c];
    sKd[r][c] = k4.x * ek; sKd[r][c + 1] = k4.y * ek;
    sKd[r][c + 2] = k4.z * ek; sKd[r][c + 3] = k4.w * ek;
#if !HAVE_ASYNC_LDS
    const float4 v4 = *(const float4*)&vp[base + (size_t)r * EH + c];
    sV[r][c] = v4.x; sV[r][c + 1] = v4.y; sV[r][c + 2] = v4.z; sV[r][c + 3] = v4.w;
    const float4 s4 = *(const float4*)&spre[soff + (size_t)r * EH + c];
    sS[r][c] = s4.x; sS[r][c + 1] = s4.y; sS[r][c + 2] = s4.z; sS[r][c + 3] = s4.w;
#endif
  }
#if HAVE_ASYNC_LDS
  WAIT_ASYNC(0);
#endif
  __syncthreads();
  const int wid = threadIdx.x >> 5;
  const int lane = threadIdx.x & 31;
  const int col = lane & 15;
  const int rb = (lane >> 4) << 3;
  // A = Qd @ Kd^T, causal mask s<=t
#pragma unroll
  for (int t = 0; t < 2; ++t) {
    const int t16 = wid * 2 + t;
    const int ti = t16 >> 2, tj = t16 & 3;
    v8f acc = zero_v8f();
    acc = wmma_k(&sQ[ti * 16][0], EH + 1, 1, &sKd[tj * 16][0], 1, EH + 1, EH, acc);
#pragma unroll
    for (int i = 0; i < 8; ++i) {
      const int tt = ti * 16 + rb + i;
      const int ss = tj * 16 + col;
      sA[tt][ss] = (ss <= tt) ? acc[i] : 0.f;
    }
  }
  __syncthreads();
  // num = A @ V + Qd @ S_prefix
  v8f num[2];
#pragma unroll
  for (int t = 0; t < 2; ++t) {
    const int t16 = wid * 2 + t;
    const int ti = t16 >> 2, tj = t16 & 3;
    v8f acc = zero_v8f();
    acc = wmma_k(&sA[ti * 16][0], CK + 1, 1, &sV[0][tj * 16], EH + 1, 1, CK, acc);
    acc = wmma_k(&sQ[ti * 16][0], EH + 1, 1, &sS[0][tj * 16], EH + 1, 1, EH, acc);
    num[t] = acc;
  }
  // den_t = 1e-6 + rowsum(A) + Qd[t].ksum_prefix
  if (threadIdx.x < CK) {
    const int tt = threadIdx.x;
    float d = 1e-6f;
    for (int ss = 0; ss < CK; ++ss) d += sA[tt][ss];
    for (int e = 0; e < EH; ++e) d += sQ[tt][e] * skp[e];
    sden[tt] = d;
  }
  __syncthreads();
#pragma unroll
  for (int t = 0; t < 2; ++t) {
    const int t16 = wid * 2 + t;
    const int ti = t16 >> 2, tj = t16 & 3;
#pragma unroll
    for (int i = 0; i < 8; ++i) {
      const int tt = ti * 16 + rb + i;
      const int f = tj * 16 + col;
      const int l = ch * CK + tt;
      y[((size_t)b * SLEN + l) * DM + h * EH + f] = num[t][i] / sden[tt];
    }
  }
}

// ---------------- Kernel 6: output projection GEMM -------------------------
__global__ __launch_bounds__(256) void k_outproj(
    const float* __restrict__ yin, const float* __restrict__ w,
    const float* __restrict__ bias, float* __restrict__ z) {
  __shared__ float lA[NBUF][64][65];
  __shared__ float lB[NBUF][64][65];
  const int row0 = blockIdx.x * 64;
  const int col0 = blockIdx.y * 64;
  const int wid = threadIdx.x >> 5;
  v8f acc[2];
  acc[0] = zero_v8f();
  acc[1] = zero_v8f();
#if HAVE_ASYNC_LDS
  for (int i = threadIdx.x; i < 64 * 16; i += 256) {
    const int r = i >> 4;
    const int c = (i & 15) << 2;
    ASYNC_CP16(&lA[0][r][c], &yin[(size_t)(row0 + r) * DM + c]);
    ASYNC_CP16(&lB[0][r][c], &w[(size_t)(col0 + r) * DM + c]);
  }
#endif
  int buf = 0;
  for (int k0 = 0; k0 < DM; k0 += 64) {
#if HAVE_ASYNC_LDS
    if (k0 + 64 < DM) {
      const int nb = buf ^ 1;
      for (int i = threadIdx.x; i < 64 * 16; i += 256) {
        const int r = i >> 4;
        const int c = (i & 15) << 2;
        ASYNC_CP16(&lA[nb][r][c], &yin[(size_t)(row0 + r) * DM + k0 + 64 + c]);
        ASYNC_CP16(&lB[nb][r][c], &w[(size_t)(col0 + r) * DM + k0 + 64 + c]);
      }
      WAIT_ASYNC(8);
    } else {
      WAIT_ASYNC(0);
    }
    __syncthreads();
#else
    __syncthreads();
    for (int i = threadIdx.x; i < 64 * 16; i += 256) {
      const int r = i >> 4;
      const int c = (i & 15) << 2;
      const float4 va = *(const float4*)&yin[(size_t)(row0 + r) * DM + k0 + c];
      lA[0][r][c] = va.x; lA[0][r][c + 1] = va.y;
      lA[0][r][c + 2] = va.z; lA[0][r][c + 3] = va.w;
      const float4 vb = *(const float4*)&w[(size_t)(col0 + r) * DM + k0 + c];
      lB[0][r][c] = vb.x; lB[0][r][c + 1] = vb.y;
      lB[0][r][c + 2] = vb.z; lB[0][r][c + 3] = vb.w;
    }
    __syncthreads();
#endif
#pragma unroll
    for (int t = 0; t < 2; ++t) {
      const int t16 = wid * 2 + t;
      const int ti = t16 >> 2, tj = t16 & 3;
      acc[t] = wmma_k(&lA[buf][ti * 16][0], 65, 1, &lB[buf][tj * 16][0], 1, 65,
                      64, acc[t]);
    }
#if HAVE_ASYNC_LDS
    __syncthreads();
    buf ^= 1;
#endif
  }
  const int lane = threadIdx.x & 31;
  const int col = lane & 15;
  const int rb = (lane >> 4) << 3;
#pragma unroll
  for (int t = 0; t < 2; ++t) {
    const int t16 = wid * 2 + t;
    const int ti = t16 >> 2, tj = t16 & 3;
#pragma unroll
    for (int i = 0; i < 8; ++i) {
      const int gr = row0 + ti * 16 + rb + i;
      const int gc = col0 + tj * 16 + col;
      z[(size_t)gr * DM + gc] = acc[t][i] + bias[gc];
    }
  }
}

// ---------------- Kernel 7: RMSNorm ----------------------------------------
__global__ __launch_bounds__(256) void k_rmsnorm(const float* __restrict__ z,
                                                 const float* __restrict__ scale,
                                                 float* __restrict__ out) {
  __shared__ float red[8];
  const int row = blockIdx.x;
  const float* zr = &z[(size_t)row * DM];
  float ss = 0.f;
  for (int i = threadIdx.x; i < DM; i += 256) {
    const float v = zr[i];
    ss += v * v;
  }
  for (int off = 16; off > 0; off >>= 1) ss += __shfl_down(ss, off, 32);
  if ((threadIdx.x & 31) == 0) red[threadIdx.x >> 5] = ss;
  __syncthreads();
  if (threadIdx.x == 0) {
    float tot = 0.f;
    for (int i = 0; i < 8; ++i) tot += red[i];
    red[0] = rsqrtf(tot / (float)DM + 1e-8f);
  }
  __syncthreads();
  const float r = red[0];
  for (int i = threadIdx.x; i < DM; i += 256)
    out[(size_t)row * DM + i] = zr[i] * r * scale[i];
}

extern "C" void kernel_launch(void* const* d_in, const int* in_sizes, int n_in,
                              void* d_out, int out_size, void* d_ws,
                              size_t ws_size, hipStream_t stream) {
  const float* x = (const float*)d_in[0];
  const float* qkv_w = (const float*)d_in[1];
  const float* qkv_b = (const float*)d_in[2];
  const float* out_w = (const float*)d_in[3];
  const float* out_b = (const float*)d_in[4];
  const float* decay_w = (const float*)d_in[5];
  const float* decay_b = (const float*)d_in[6];
  const float* norm_scale = (const float*)d_in[7];
  float* out = (float*)d_out;
  float* ws = (float*)d_ws;

  const size_t NTE = (size_t)NB * NH * SLEN * EH; // 4,194,304 floats
  float* qp = ws;               // q' (b,h,l,e)
  float* kp = qp + NTE;         // k'
  float* vp = kp + NTE;         // v
  float* sloc = vp + NTE;       // chunk states -> (in-place) exclusive prefixes
  float* y = sloc + NTE;        // attention output (b,l,d)
  float* cumul = y + NTE;       // (b,h,l) chunk-relative log-lambda cumsum
  float* csum = cumul + (size_t)NB * NH * SLEN; // (b,h,ch) chunk totals
  float* ksl = csum + (size_t)NB * NH * NCH;    // ksum local -> prefixes
  float* z = qp;                // reuse qp for pre-norm projection output
  // total ws: ~21.1M floats (~84.5 MB)

  k_qkv<<<dim3(NTOK / 64, D3 / 64), 256, 0, stream>>>(x, qkv_w, qkv_b, qp, kp, vp);
  k_decay<<<dim3(NB * NCH), 64, 0, stream>>>(x, decay_w, decay_b, cumul, csum);
  k_chunk_local<<<dim3(NB * NH * NCH), 256, 0, stream>>>(kp, vp, cumul, csum,
                                                         sloc, ksl);
  k_scan<<<dim3(NB * NH), 256, 0, stream>>>(sloc, ksl, csum);
  k_chunk_out<<<dim3(NB * NH * NCH), 256, 0, stream>>>(qp, kp, vp, cumul, sloc,
                                                       ksl, y);
  k_outproj<<<dim3(NTOK / 64, DM / 64), 256, 0, stream>>>(y, out_w, out_b, z);
  k_rmsnorm<<<dim3(NTOK), 256, 0, stream>>>(z, norm_scale, out);
}